// PairwiseDistanceLoss_38465727103844
// MI455X (gfx1250) — compile-verified
//
#include <hip/hip_runtime.h>
#include <hip/hip_bf16.h>

typedef float v2f __attribute__((ext_vector_type(2)));
typedef float v8f __attribute__((ext_vector_type(8)));

#define EPS 1e-8f

constexpr int BATCH   = 4;
constexpr int NPTS    = 4096;
constexpr int TILES   = NPTS / 16;       // 256 column tiles
constexpr int RBLOCKS = NPTS / 16;       // 256 row blocks
constexpr int WGS     = BATCH * RBLOCKS; // 1024 workgroups
constexpr int WAVES   = 8;               // 256 threads / wave32
// ws layout (32-bit slots):
//   [0..3]                 max(clip(dsqP)+EPS) bits per batch (atomicMax, order-invariant)
//   [4..7]                 max(clip(dsqT)+EPS) bits per batch
//   [16          .. +WGS)  per-WG partial  S_aa = sum(clipP+EPS)
//   [16 +   WGS  .. +WGS)  per-WG partial  S_bb = sum(clipT+EPS)
//   [16 + 2*WGS  .. +WGS)  per-WG partial  S_ab = sum(sqrt((clipP+EPS)*(clipT+EPS)))
constexpr size_t SMEM_BYTES = (size_t)2 * NPTS * sizeof(float4) + WAVES * 8 * sizeof(float);

extern __shared__ unsigned char cdna5_smem[];

__device__ __forceinline__ float clip0(float x) {
    // single v_med3_f32: median(x, 0, +inf) == max(x, 0) for finite x,
    // no NaN-canonicalization pre-op (WMMA outputs are finite here).
    return __builtin_amdgcn_fmed3f(x, 0.0f, __builtin_inff());
}

__device__ __forceinline__ float wave_max(float v) {
#pragma unroll
    for (int off = 16; off > 0; off >>= 1)
        v = fmaxf(v, __shfl_xor(v, off, 32));
    return v;
}

__device__ __forceinline__ float wave_sum(float v) {
#pragma unroll
    for (int off = 16; off > 0; off >>= 1)
        v += __shfl_xor(v, off, 32);
    return v;
}

__global__ void init_ws(unsigned int* __restrict__ wsMax) {
    if (threadIdx.x < 8) wsMax[threadIdx.x] = 0u;
}

// One fused pass: per (batch, 16-row block) workgroup, sweep all 256 column tiles.
// dist^2 tile = one V_WMMA_F32_16X16X4_F32 via homogeneous K=4 trick:
//   A row m = (x,y,z,1), B col n = (-2x,-2y,-2z,||p_n||^2), C[m][n] = ||p_m||^2.
__global__ void __launch_bounds__(256)
fused_pass(const float* __restrict__ pred, const float* __restrict__ target,
           unsigned int* __restrict__ wsMax,
           float* __restrict__ pAA, float* __restrict__ pBB, float* __restrict__ pAB) {
    float4* sP  = (float4*)cdna5_smem;          // staged B-columns, pred
    float4* sT  = sP + NPTS;                    // staged B-columns, target
    float*  red = (float*)(sT + NPTS);          // cross-wave reduction scratch (8 waves x 5)

    const int wg      = blockIdx.x;
    const int batch   = wg >> 8;
    const int rowBase = (wg & 255) * 16;
    const int lane    = threadIdx.x & 31;
    const int wave    = threadIdx.x >> 5;
    const int l16     = lane & 15;
    const bool hi     = lane >= 16;

    const float* P = pred   + batch * NPTS * 3;
    const float* T = target + batch * NPTS * 3;

    // ---- Stage all 4096 points of both tensors into LDS as (-2x,-2y,-2z,||p||^2) ----
    for (int i = threadIdx.x; i < NPTS; i += 256) {     // uniform trip count (16)
        const float* q = P + 3 * i;
        float x = q[0], y = q[1], z = q[2];
        sP[i] = make_float4(-2.0f * x, -2.0f * y, -2.0f * z, x * x + y * y + z * z);
        const float* r = T + 3 * i;
        x = r[0]; y = r[1]; z = r[2];
        sT[i] = make_float4(-2.0f * x, -2.0f * y, -2.0f * z, x * x + y * y + z * z);
    }
    __syncthreads();

    // ---- Build A (16x4) and C (16x16) fragments for this row block, from LDS ----
    float4 smP = sP[rowBase + l16];
    float4 smT = sT[rowBase + l16];
    v2f aP, aT;                                          // A layout: lanes0-15 (x,y); lanes16-31 (z,1)
    aP.x = hi ? -0.5f * smP.z : -0.5f * smP.x;
    aP.y = hi ? 1.0f          : -0.5f * smP.y;
    aT.x = hi ? -0.5f * smT.z : -0.5f * smT.x;
    aT.y = hi ? 1.0f          : -0.5f * smT.y;

    const int rb = rowBase + (hi ? 8 : 0);               // C layout: row = vgpr + 8*(lane>=16)
    v8f cP, cT;
#pragma unroll
    for (int v = 0; v < 8; ++v) {
        cP[v] = sP[rb + v].w;
        cT[v] = sT[rb + v].w;
    }

    // ---- Sweep column tiles: 2 WMMAs + 2 conflict-free ds_load_b64 per tile ----
    const int off2 = hi ? 2 : 0;                         // B layout: lanes0-15 (-2x,-2y); lanes16-31 (-2z, sq)
    float aa = 0.0f, bb = 0.0f, ab = 0.0f, mP = 0.0f, mT = 0.0f;
    for (int J = wave; J < TILES; J += WAVES) {          // uniform trip count (32) per wave
        const int n = J * 16 + l16;
        v2f bP = *(const v2f*)((const float*)(sP + n) + off2);
        v2f bT = *(const v2f*)((const float*)(sT + n) + off2);
        v8f dP = __builtin_amdgcn_wmma_f32_16x16x4_f32(false, aP, false, bP,
                                                       (short)0, cP, false, false);
        v8f dT = __builtin_amdgcn_wmma_f32_16x16x4_f32(false, aT, false, bT,
                                                       (short)0, cT, false, false);
#pragma unroll
        for (int v = 0; v < 8; ++v) {
            float ap2 = clip0(dP[v]) + EPS;              // = a^2 (clip + eps), one v_med3
            float bt2 = clip0(dT[v]) + EPS;              // = b^2
            aa += ap2;
            bb += bt2;
            // raw v_sqrt_f32 (TRANS, co-executes): args in [1e-16, ~4e3], far from
            // the denormal range the libm fixup sequence guards against.
            ab += __builtin_amdgcn_sqrtf(ap2 * bt2);     // a*b: one sqrt per element
            mP = fmaxf(mP, ap2);
            mT = fmaxf(mT, bt2);
        }
    }

    // ---- Reductions: fixed shfl tree, then fixed-order cross-wave in LDS ----
    aa = wave_sum(aa); bb = wave_sum(bb); ab = wave_sum(ab);
    mP = wave_max(mP); mT = wave_max(mT);
    if (lane == 0) {
        float* w = red + wave * 5;
        w[0] = aa; w[1] = bb; w[2] = ab; w[3] = mP; w[4] = mT;
    }
    __syncthreads();
    if (threadIdx.x == 0) {
        float taa = 0.0f, tbb = 0.0f, tab = 0.0f, tmP = 0.0f, tmT = 0.0f;
        for (int w = 0; w < WAVES; ++w) {                // fixed order: deterministic
            const float* r = red + w * 5;
            taa += r[0]; tbb += r[1]; tab += r[2];
            tmP = fmaxf(tmP, r[3]); tmT = fmaxf(tmT, r[4]);
        }
        pAA[wg] = taa; pBB[wg] = tbb; pAB[wg] = tab;
        atomicMax(&wsMax[batch],     __float_as_uint(tmP));  // values > 0: uint order == float order
        atomicMax(&wsMax[4 + batch], __float_as_uint(tmT));
    }
}

// Combine per-batch: sum_diff_b = invP^2*Saa + invT^2*Sbb - 2*invP*invT*Sab (double).
__global__ void __launch_bounds__(256)
finalize(const unsigned int* __restrict__ wsMax,
         const float* __restrict__ pAA, const float* __restrict__ pBB,
         const float* __restrict__ pAB, float* __restrict__ out) {
    __shared__ double s0[256], s1[256], s2[256];
    const int tid = threadIdx.x;
    double total = 0.0;
    for (int b = 0; b < BATCH; ++b) {
        const int base = b * RBLOCKS;                    // 256 partials per batch
        s0[tid] = (double)pAA[base + tid];
        s1[tid] = (double)pBB[base + tid];
        s2[tid] = (double)pAB[base + tid];
        __syncthreads();
#pragma unroll
        for (int off = 128; off > 0; off >>= 1) {        // fixed tree: deterministic
            if (tid < off) {
                s0[tid] += s0[tid + off];
                s1[tid] += s1[tid + off];
                s2[tid] += s2[tid + off];
            }
            __syncthreads();
        }
        if (tid == 0) {
            // stored max = max(clip)+EPS, so max(dist) = sqrt(stored)
            double invP = 1.0 / (sqrt((double)__uint_as_float(wsMax[b]))     + (double)EPS);
            double invT = 1.0 / (sqrt((double)__uint_as_float(wsMax[4 + b])) + (double)EPS);
            total += invP * invP * s0[0] + invT * invT * s1[0] - 2.0 * invP * invT * s2[0];
        }
        __syncthreads();
    }
    if (tid == 0) {
        const double count = (double)BATCH * (double)NPTS * (double)NPTS;
        out[0] = (float)sqrt(total / count + (double)EPS);
    }
}

extern "C" void kernel_launch(void* const* d_in, const int* in_sizes, int n_in,
                              void* d_out, int out_size, void* d_ws, size_t ws_size,
                              hipStream_t stream) {
    (void)in_sizes; (void)n_in; (void)out_size; (void)ws_size;
    const float* pred   = (const float*)d_in[0];
    const float* target = (const float*)d_in[1];
    unsigned int* wsMax = (unsigned int*)d_ws;
    float* pAA = (float*)d_ws + 16;
    float* pBB = pAA + WGS;
    float* pAB = pBB + WGS;
    float* out = (float*)d_out;

    init_ws<<<1, 32, 0, stream>>>(wsMax);
    fused_pass<<<WGS, 256, SMEM_BYTES, stream>>>(pred, target, wsMax, pAA, pBB, pAB);
    finalize<<<1, 256, 0, stream>>>(wsMax, pAA, pBB, pAB, out);
}